// OWL_prox_58497454571816
// MI455X (gfx1250) — compile-verified
//
#include <hip/hip_runtime.h>
#include <hip/hip_bf16.h>

typedef __attribute__((ext_vector_type(2))) float v2f;
typedef __attribute__((ext_vector_type(8))) float v8f;

#define CHSZ 1024

// ---------------- kernel 0: build sort keys ----------------
// key = ~bits(|u-x|): abs values are >= 0 so IEEE bits are monotone in value;
// ascending radix sort on ~bits == stable descending sort on |diff|.
__global__ __launch_bounds__(256) void k_init(const float* __restrict__ u,
                                              const float* __restrict__ x,
                                              unsigned* __restrict__ key,
                                              unsigned* __restrict__ idx, int n) {
  int i = blockIdx.x * blockDim.x + threadIdx.x;
  if (i >= n) return;
  float d = u[i] - x[i];
  float a = fabsf(d);
  key[i] = ~__float_as_uint(a);
  idx[i] = (unsigned)i;
}

// wave32 match-any over an 8-bit digit via 8 ballots
__device__ inline unsigned long long match_digit(unsigned digit,
                                                 unsigned long long active_mask) {
  unsigned long long mask = active_mask;
#pragma unroll
  for (int b = 0; b < 8; ++b) {
    unsigned long long vote = __ballot((digit >> b) & 1u);
    mask &= ((digit >> b) & 1u) ? vote : ~vote;
  }
  return mask & active_mask;
}

// ---------------- radix pass: per-block digit histogram ----------------
__global__ __launch_bounds__(256) void k_hist(const unsigned* __restrict__ key,
                                              unsigned* __restrict__ hist,
                                              int nb, int n, int shift) {
  __shared__ unsigned wh[8 * 256];
  int t = threadIdx.x;
#pragma unroll
  for (int w = 0; w < 8; ++w) wh[w * 256 + t] = 0;
  __syncthreads();
  int i = blockIdx.x * 256 + t;
  bool active = i < n;
  unsigned digit = active ? ((key[i] >> shift) & 255u) : 0u;
  unsigned lane = (unsigned)t & 31u;
  unsigned wave = (unsigned)t >> 5;
  unsigned long long am = __ballot(active ? 1 : 0);
  unsigned long long mask = match_digit(digit, am);
  if (active && lane == (unsigned)(__ffsll((unsigned long long)mask) - 1)) {
    wh[wave * 256 + digit] = (unsigned)__popcll(mask);
  }
  __syncthreads();
  unsigned sum = 0;
#pragma unroll
  for (int w = 0; w < 8; ++w) sum += wh[w * 256 + t];
  // digit-major layout: all blocks of digit d contiguous -> linear exclusive
  // scan of this array yields global scatter bases.
  hist[(size_t)t * (size_t)nb + (size_t)blockIdx.x] = sum;
}

// ---------------- WMMA-based exclusive scan over hist ----------------
// One wave. Each 256-value tile, viewed as a 16x16 row-major matrix V, has
// exclusive scan P = V*U_strict + L_strict*V*J  (within-row prefix plus the
// total of all previous rows). Each 16x16x16 product = 4 chained
// V_WMMA_F32_16X16X4_F32 (K=4). Counts < 2^24 so f32 arithmetic is exact.
__global__ __launch_bounds__(32) void k_scan_wmma(unsigned* __restrict__ hist,
                                                  int tiles) {
  __shared__ float v[256];
  __shared__ float t2[256];
  unsigned lane = threadIdx.x;
  bool lo = lane < 16;
  int Mrow = lo ? (int)lane : (int)lane - 16;  // M for A-layout, N for B-layout
  unsigned carry = 0;
  for (int tile = 0; tile < tiles; ++tile) {
    size_t base = (size_t)tile * 256;
#pragma unroll
    for (int q = 0; q < 8; ++q)
      v[lane * 8 + q] = (float)hist[base + lane * 8 + q];
    __syncthreads();

    v8f c1 = {};  // V * U_strict  (row-wise exclusive prefix)
    v8f tt = {};  // L_strict * V  (partial column sums over previous rows)
#pragma unroll
    for (int s = 0; s < 4; ++s) {
      // 32-bit A 16x4 layout: vgpr0 = {K=0 | K=2}, vgpr1 = {K=1 | K=3};
      // B 4x16 mirrors it with lanes carrying N.
      int k0 = 4 * s + (lo ? 0 : 2);
      int k1 = k0 + 1;
      v2f va; va.x = v[Mrow * 16 + k0]; va.y = v[Mrow * 16 + k1];      // A = V
      v2f vb; vb.x = v[k0 * 16 + Mrow]; vb.y = v[k1 * 16 + Mrow];      // B = V
      // U_strict (B-layout, B[k][N] = k<N) == L_strict (A-layout, A[M][K] = K<M)
      v2f tri; tri.x = (k0 < Mrow) ? 1.0f : 0.0f;
               tri.y = (k1 < Mrow) ? 1.0f : 0.0f;
      c1 = __builtin_amdgcn_wmma_f32_16x16x4_f32(false, va, false, tri,
                                                 (short)0, c1, false, false);
      tt = __builtin_amdgcn_wmma_f32_16x16x4_f32(false, tri, false, vb,
                                                 (short)0, tt, false, false);
    }

    // D-layout -> linear LDS so tt can be re-fed as an A operand
#pragma unroll
    for (int j = 0; j < 8; ++j) {
      int m = lo ? j : j + 8;
      int nn = lo ? (int)lane : (int)lane - 16;
      t2[m * 16 + nn] = tt[j];
    }
    __syncthreads();

    v8f d = c1;
    v2f ones; ones.x = 1.0f; ones.y = 1.0f;   // J (all-ones) in B-layout
#pragma unroll
    for (int s = 0; s < 4; ++s) {
      int k0 = 4 * s + (lo ? 0 : 2);
      v2f ta; ta.x = t2[Mrow * 16 + k0]; ta.y = t2[Mrow * 16 + k0 + 1];
      d = __builtin_amdgcn_wmma_f32_16x16x4_f32(false, ta, false, ones,
                                                (short)0, d, false, false);
    }

    // tile total = exclusive[255] + v[255]; element 255 is lane31, vgpr7
    float d7 = d[7];
    float last = __shfl(d7, 31, 32);
    unsigned total = (unsigned)last + (unsigned)v[255];
#pragma unroll
    for (int j = 0; j < 8; ++j) {
      int m = lo ? j : j + 8;
      int nn = lo ? (int)lane : (int)lane - 16;
      hist[base + (size_t)(m * 16 + nn)] = carry + (unsigned)d[j];
    }
    carry += total;
    __syncthreads();
  }
}

// ---------------- radix pass: stable scatter ----------------
__global__ __launch_bounds__(256) void k_scatter(const unsigned* __restrict__ srcK,
                                                 const unsigned* __restrict__ srcI,
                                                 unsigned* __restrict__ dstK,
                                                 unsigned* __restrict__ dstI,
                                                 const unsigned* __restrict__ scan,
                                                 int nb, int n, int shift) {
  __shared__ unsigned wh[8 * 256];
  int t = threadIdx.x;
#pragma unroll
  for (int w = 0; w < 8; ++w) wh[w * 256 + t] = 0;
  __syncthreads();
  int i = blockIdx.x * 256 + t;
  bool active = i < n;
  unsigned k = active ? srcK[i] : 0u;
  unsigned vv = active ? srcI[i] : 0u;
  unsigned digit = active ? ((k >> shift) & 255u) : 0u;
  unsigned lane = (unsigned)t & 31u;
  unsigned wave = (unsigned)t >> 5;
  unsigned long long am = __ballot(active ? 1 : 0);
  unsigned long long mask = match_digit(digit, am);
  unsigned rank = (unsigned)__popcll(mask & ((1ull << lane) - 1ull));
  if (active && lane == (unsigned)(__ffsll((unsigned long long)mask) - 1)) {
    wh[wave * 256 + digit] = (unsigned)__popcll(mask);
  }
  __syncthreads();
  if (active) {
    unsigned pre = 0;
    for (unsigned w = 0; w < wave; ++w) pre += wh[w * 256 + digit];
    unsigned pos = scan[(size_t)digit * (size_t)nb + (size_t)blockIdx.x] + pre + rank;
    dstK[pos] = k;
    dstI[pos] = vv;
  }
}

// ---------------- chunked PAVA (non-increasing fit of y, strict-> merge) ----
// Matches the reference's cross-multiplied violation test on z=-y exactly
// (sign negation is exact): merge while sum[t-1]*cnt[t-2] > sum[t-2]*cnt[t-1].
__global__ void k_chunk_pava(const unsigned* __restrict__ keyS,
                             const float* __restrict__ w,
                             float* __restrict__ psum, float* __restrict__ pcnt,
                             unsigned* __restrict__ chunkCnt, int n, int nchunks) {
  int c = blockIdx.x * blockDim.x + threadIdx.x;
  if (c >= nchunks) return;
  int base = c * CHSZ;
  int end = base + CHSZ;
  if (end > n) end = n;
  int top = 0;
  for (int r = base; r < end; ++r) {
    float a = __uint_as_float(~keyS[r]);   // sorted |diff| at rank r
    float y = a - w[r];
    psum[base + top] = y;
    pcnt[base + top] = 1.0f;
    ++top;
    while (top >= 2) {
      float s1 = psum[base + top - 1], c1 = pcnt[base + top - 1];
      float s0 = psum[base + top - 2], c0 = pcnt[base + top - 2];
      if (!(s1 * c0 > s0 * c1)) break;
      psum[base + top - 2] = s0 + s1;
      pcnt[base + top - 2] = c0 + c1;
      --top;
    }
  }
  chunkCnt[c] = (unsigned)top;
}

// ---------------- sequential merge of chunk pool stacks ----------------
__global__ void k_merge(const float* __restrict__ psum, const float* __restrict__ pcnt,
                        const unsigned* __restrict__ chunkCnt,
                        float* __restrict__ gsum, float* __restrict__ gcnt,
                        float* __restrict__ poolMean, unsigned* __restrict__ poolEnds,
                        unsigned* __restrict__ gcount, int nchunks) {
  if (threadIdx.x != 0 || blockIdx.x != 0) return;
  int gt = 0;
  for (int c = 0; c < nchunks; ++c) {
    int base = c * CHSZ;
    int pc = (int)chunkCnt[c];
    for (int pp = 0; pp < pc; ++pp) {
      gsum[gt] = psum[base + pp];
      gcnt[gt] = pcnt[base + pp];
      ++gt;
      while (gt >= 2) {
        float s1 = gsum[gt - 1], c1 = gcnt[gt - 1];
        float s0 = gsum[gt - 2], c0 = gcnt[gt - 2];
        if (!(s1 * c0 > s0 * c1)) break;
        gsum[gt - 2] = s0 + s1;
        gcnt[gt - 2] = c0 + c1;
        --gt;
      }
    }
  }
  unsigned acc = 0;
  for (int g = 0; g < gt; ++g) {
    float m = gsum[g] / gcnt[g];
    poolMean[g] = m > 0.0f ? m : 0.0f;     // clip at 0 (y_min = 0)
    acc += (unsigned)gcnt[g];
    poolEnds[g] = acc;
  }
  *gcount = (unsigned)gt;
}

// ---------------- fill fitted values + scatter through inverse perm --------
__global__ __launch_bounds__(256) void k_final(const unsigned* __restrict__ idxS,
                                               const unsigned* __restrict__ poolEnds,
                                               const float* __restrict__ poolMean,
                                               const unsigned* __restrict__ gcount,
                                               const float* __restrict__ u,
                                               const float* __restrict__ x,
                                               float* __restrict__ out, int n) {
  __builtin_prefetch(poolEnds, 0, 3);  // global_prefetch_b8 (gfx1250)
  int r = blockIdx.x * blockDim.x + threadIdx.x;
  if (r >= n) return;
  int G = (int)*gcount;
  int lo = 0, hi = G - 1;
  while (lo < hi) {  // first g with poolEnds[g] > r (== searchsorted 'right')
    int mid = (lo + hi) >> 1;
    if (poolEnds[mid] > (unsigned)r) hi = mid; else lo = mid + 1;
  }
  float fv = poolMean[lo];
  unsigned i = idxS[r];
  float d = u[i] - x[i];
  float s = (d > 0.0f) ? 1.0f : ((d < 0.0f) ? -1.0f : 0.0f);
  out[i] = x[i] + s * fv;
}

extern "C" void kernel_launch(void* const* d_in, const int* in_sizes, int n_in,
                              void* d_out, int out_size, void* d_ws, size_t ws_size,
                              hipStream_t stream) {
  const float* u = (const float*)d_in[0];
  const float* x = (const float*)d_in[1];
  const float* w = (const float*)d_in[2];
  float* out = (float*)d_out;
  int n = in_sizes[0];
  int nb = (n + 255) / 256;
  int histN = nb * 256;
  int nchunks = (n + CHSZ - 1) / CHSZ;

  char* base = (char*)d_ws;
  size_t off = 0;
  auto take = [&](size_t bytes) -> void* {
    void* r = base + off;
    off += (bytes + 255) & ~(size_t)255;
    return r;
  };
  unsigned* gcount   = (unsigned*)take(sizeof(unsigned));
  unsigned* chunkCnt = (unsigned*)take((size_t)nchunks * 4);
  unsigned* keyA     = (unsigned*)take((size_t)n * 4);
  unsigned* idxA     = (unsigned*)take((size_t)n * 4);
  unsigned* keyB     = (unsigned*)take((size_t)n * 4);
  unsigned* idxB     = (unsigned*)take((size_t)n * 4);
  unsigned* hist     = (unsigned*)take((size_t)histN * 4);
  float*    psum     = (float*)take((size_t)n * 4);
  float*    pcnt     = (float*)take((size_t)n * 4);
  float*    gsum     = (float*)take((size_t)n * 4);
  float*    gcnt     = (float*)take((size_t)n * 4);
  float*    poolMean = (float*)take((size_t)n * 4);
  unsigned* poolEnds = (unsigned*)take((size_t)n * 4);
  (void)ws_size; (void)n_in; (void)out_size;

  k_init<<<nb, 256, 0, stream>>>(u, x, keyA, idxA, n);

  unsigned* sK = keyA; unsigned* sI = idxA;
  unsigned* dK = keyB; unsigned* dI = idxB;
  for (int pass = 0; pass < 4; ++pass) {
    int shift = pass * 8;
    k_hist<<<nb, 256, 0, stream>>>(sK, hist, nb, n, shift);
    k_scan_wmma<<<1, 32, 0, stream>>>(hist, nb);
    k_scatter<<<nb, 256, 0, stream>>>(sK, sI, dK, dI, hist, nb, n, shift);
    unsigned* tK = sK; sK = dK; dK = tK;
    unsigned* tI = sI; sI = dI; dI = tI;
  }
  // after 4 passes the sorted keys/indices are back in keyA/idxA (== sK/sI)

  int cblocks = (nchunks + 63) / 64;
  k_chunk_pava<<<cblocks, 64, 0, stream>>>(sK, w, psum, pcnt, chunkCnt, n, nchunks);
  k_merge<<<1, 1, 0, stream>>>(psum, pcnt, chunkCnt, gsum, gcnt, poolMean, poolEnds,
                               gcount, nchunks);
  k_final<<<nb, 256, 0, stream>>>(sI, poolEnds, poolMean, gcount, u, x, out, n);
}